// Gelato_51634096833002
// MI455X (gfx1250) — compile-verified
//
#include <hip/hip_runtime.h>
#include <math.h>

typedef __attribute__((ext_vector_type(2))) float v2f;
typedef __attribute__((ext_vector_type(8))) float v8f;

__device__ __forceinline__ v8f wmma_f32_16x16x4(v2f a, v2f b, v8f c) {
  // D = A(16x4 f32) x B(4x16 f32) + C(16x16 f32), wave32
  return __builtin_amdgcn_wmma_f32_16x16x4_f32(
      /*neg_a=*/false, a, /*neg_b=*/false, b,
      /*c_mod=*/(short)0, c, /*reuse_a=*/false, /*reuse_b=*/false);
}

// ---------------------------------------------------------------------------
// K1: per-edge MLP -> vals[e] = softmax(h @ W3 + b3)[:,1]
// One wave per edge; ex/h staged in per-wave LDS slices.
// ---------------------------------------------------------------------------
__global__ __launch_bounds__(256) void mlp_kernel(
    const float* __restrict__ X, const long long* __restrict__ aug_edges, int E,
    const float* __restrict__ W1, const float* __restrict__ b1,
    const float* __restrict__ W2, const float* __restrict__ b2,
    const float* __restrict__ W3, const float* __restrict__ b3,
    float* __restrict__ vals)
{
  __shared__ float sEx[8][256];
  __shared__ float sH[8][128];
  const int wid  = threadIdx.x >> 5;
  const int lane = threadIdx.x & 31;
  const int e_raw = blockIdx.x * 8 + wid;
  const int e = (e_raw < E) ? e_raw : (E - 1);   // clamp so all waves stay alive

  const long long i0 = aug_edges[2 * (size_t)e + 0];
  const long long i1 = aug_edges[2 * (size_t)e + 1];
  const float* x0 = X + (size_t)i0 * 128;
  const float* x1 = X + (size_t)i1 * 128;

  #pragma unroll
  for (int t = 0; t < 4; ++t) {
    int k = lane + 32 * t;
    float a = x0[k], b = x1[k];
    sEx[wid][k]       = a + b;
    sEx[wid][128 + k] = fabsf(a - b);
  }
  __syncthreads();

  float acc[4];
  #pragma unroll
  for (int t = 0; t < 4; ++t) acc[t] = b1[lane + 32 * t];
  for (int k = 0; k < 256; ++k) {
    float ek = sEx[wid][k];
    #pragma unroll
    for (int t = 0; t < 4; ++t) acc[t] += ek * W1[k * 128 + lane + 32 * t];
  }
  #pragma unroll
  for (int t = 0; t < 4; ++t) sH[wid][lane + 32 * t] = fmaxf(acc[t], 0.0f);
  __syncthreads();

  #pragma unroll
  for (int t = 0; t < 4; ++t) acc[t] = b2[lane + 32 * t];
  for (int k = 0; k < 128; ++k) {
    float hk = sH[wid][k];
    #pragma unroll
    for (int t = 0; t < 4; ++t) acc[t] += hk * W2[k * 128 + lane + 32 * t];
  }
  __syncthreads();
  #pragma unroll
  for (int t = 0; t < 4; ++t) sH[wid][lane + 32 * t] = fmaxf(acc[t], 0.0f);
  __syncthreads();

  float z0 = 0.f, z1 = 0.f;
  #pragma unroll
  for (int t = 0; t < 4; ++t) {
    int k = lane + 32 * t;
    float hk = sH[wid][k];
    z0 += hk * W3[2 * k + 0];
    z1 += hk * W3[2 * k + 1];
  }
  #pragma unroll
  for (int off = 16; off > 0; off >>= 1) {
    z0 += __shfl_xor(z0, off, 32);
    z1 += __shfl_xor(z1, off, 32);
  }
  if (lane == 0 && e_raw < E) {
    float a0 = z0 + b3[0], a1 = z1 + b3[1];
    vals[e_raw] = 1.0f / (1.0f + __expf(a0 - a1));   // softmax[:,1] of 2 logits
  }
}

// ---------------------------------------------------------------------------
// K2: A_enh base = 0.5*A + 0.25*mask*S, diag = 1
// ---------------------------------------------------------------------------
__global__ __launch_bounds__(256) void build_kernel(
    const float* __restrict__ A, const float* __restrict__ S,
    const float* __restrict__ mask, float* __restrict__ M, int n)
{
  size_t idx = (size_t)blockIdx.x * 256 + threadIdx.x;
  size_t nn = (size_t)n * n;
  if (idx >= nn) return;
  int i = (int)(idx / (size_t)n);
  int j = (int)(idx - (size_t)i * n);
  M[idx] = (i == j) ? 1.0f : (0.5f * A[idx] + 0.25f * mask[idx] * S[idx]);
}

// ---------------------------------------------------------------------------
// K3: symmetric scatter of 0.25*vals (aug_edges are unique triu positions)
// ---------------------------------------------------------------------------
__global__ __launch_bounds__(256) void scatter_kernel(
    const long long* __restrict__ aug_edges, const float* __restrict__ vals,
    float* __restrict__ M, int E, int n)
{
  int e = blockIdx.x * 256 + threadIdx.x;
  if (e >= E) return;
  long long a = aug_edges[2 * (size_t)e + 0];
  long long b = aug_edges[2 * (size_t)e + 1];
  float v = 0.25f * vals[e];
  M[(size_t)a * n + b] += v;
  M[(size_t)b * n + a] += v;
}

// ---------------------------------------------------------------------------
// K4: row sums d[i]
// ---------------------------------------------------------------------------
__global__ __launch_bounds__(256) void rowsum_kernel(
    const float* __restrict__ M, float* __restrict__ d, int n)
{
  __shared__ float red[256];
  int i = blockIdx.x;
  const float* row = M + (size_t)i * n;
  float s = 0.f;
  for (int j = threadIdx.x; j < n; j += 256) s += row[j];
  red[threadIdx.x] = s;
  __syncthreads();
  for (int o = 128; o > 0; o >>= 1) {
    if (threadIdx.x < o) red[threadIdx.x] += red[threadIdx.x + o];
    __syncthreads();
  }
  if (threadIdx.x == 0) d[i] = red[0];
}

// ---------------------------------------------------------------------------
// K5: pi = d / sum(|d|)   (single block, deterministic tree)
// ---------------------------------------------------------------------------
__global__ __launch_bounds__(1024) void pi_kernel(
    const float* __restrict__ d, float* __restrict__ pi, int n)
{
  __shared__ float red[1024];
  float s = 0.f;
  for (int i = threadIdx.x; i < n; i += 1024) s += fabsf(d[i]);
  red[threadIdx.x] = s;
  __syncthreads();
  for (int o = 512; o > 0; o >>= 1) {
    if (threadIdx.x < o) red[threadIdx.x] += red[threadIdx.x + o];
    __syncthreads();
  }
  float inv = 1.0f / red[0];
  for (int i = threadIdx.x; i < n; i += 1024) pi[i] = d[i] * inv;
}

// ---------------------------------------------------------------------------
// K6: M = A_enh / d[i]  (row-normalize in place)
// ---------------------------------------------------------------------------
__global__ __launch_bounds__(256) void normalize_kernel(
    float* __restrict__ M, const float* __restrict__ d, int n)
{
  size_t idx = (size_t)blockIdx.x * 256 + threadIdx.x;
  size_t nn = (size_t)n * n;
  if (idx >= nn) return;
  int i = (int)(idx / (size_t)n);
  M[idx] = M[idx] / d[i];
}

// ---------------------------------------------------------------------------
// K7: P = M @ M via V_WMMA_F32_16X16X4_F32, fused R-statistics epilogue.
// 128x128 block tile, 8 waves (4x2), each wave 32x64 = 2x4 WMMA accumulators.
// LDS tiles in K-paired layout: tile[kpair][row][2] so each WMMA fragment is
// one aligned ds_load_b64 straight into an even VGPR pair (no repack movs).
// Row stride 288 floats => lanes 0-15 (kpair) and 16-31 (kpair+1) hit
// disjoint 32-bank halves: conflict-free b64 reads.
// Writes deterministic per-block double partials (sum R, sum R^2).
// ---------------------------------------------------------------------------
#define KP_STRIDE 288   // 128*2 data floats + 32 pad floats

__global__ __launch_bounds__(256) void gemm_stats_kernel(
    const float* __restrict__ M, const float* __restrict__ pi,
    double* __restrict__ partials, int n)
{
  __shared__ float AsF[16 * KP_STRIDE];   // [kpair][m][2], kpair = k/2
  __shared__ float BsF[16 * KP_STRIDE];   // [kpair][c][2]
  __shared__ double wred[16];

  const int wid  = threadIdx.x >> 5;
  const int lane = threadIdx.x & 31;
  const int wm = wid & 3;    // row group (32 rows each)
  const int wn = wid >> 2;   // col group (64 cols each)
  const int bm = blockIdx.y, bn = blockIdx.x;

  const v8f vzero = {0.f, 0.f, 0.f, 0.f, 0.f, 0.f, 0.f, 0.f};
  v8f acc[2][4];
  #pragma unroll
  for (int mt = 0; mt < 2; ++mt)
    #pragma unroll
    for (int nt = 0; nt < 4; ++nt) acc[mt][nt] = vzero;

  const size_t rowBase = (size_t)bm * 128;
  const size_t colBase = (size_t)bn * 128;
  const int kpHalf = lane >> 4;            // lanes 16-31 read kpair+1
  const int mrow   = lane & 15;
  const int mA0 = wm * 32 + mrow;          // fragment row, mt adds 16
  const int cB0 = wn * 64 + mrow;          // fragment col, nt adds 16

  for (int k0 = 0; k0 < n; k0 += 32) {
    // ---- stage A tile: 128 rows x 32 k, float4 loads, K-paired stores ----
    {
      int t = threadIdx.x;
      #pragma unroll
      for (int it = 0; it < 4; ++it, t += 256) {       // 1024 float4 total
        int r  = t >> 3;                                // 0..127
        int c4 = (t & 7) << 2;                          // 0,4,...,28
        float4 v = *(const float4*)&M[(rowBase + r) * (size_t)n + (k0 + c4)];
        float* p0 = &AsF[((c4 >> 1) + 0) * KP_STRIDE + 2 * r];
        float* p1 = &AsF[((c4 >> 1) + 1) * KP_STRIDE + 2 * r];
        ((v2f*)p0)[0] = (v2f){v.x, v.y};
        ((v2f*)p1)[0] = (v2f){v.z, v.w};
      }
    }
    // ---- stage B tile: 32 k x 128 cols, float4 loads, K-paired stores ----
    {
      int t = threadIdx.x;
      #pragma unroll
      for (int it = 0; it < 4; ++it, t += 256) {       // 1024 float4 total
        int r  = t >> 5;                                // 0..31 (k)
        int c4 = (t & 31) << 2;                         // 0,4,...,124
        float4 v = *(const float4*)&M[(size_t)(k0 + r) * n + colBase + c4];
        float* p = &BsF[(r >> 1) * KP_STRIDE + (r & 1)];
        p[2 * (c4 + 0)] = v.x;
        p[2 * (c4 + 1)] = v.y;
        p[2 * (c4 + 2)] = v.z;
        p[2 * (c4 + 3)] = v.w;
      }
    }
    __syncthreads();

    #pragma unroll
    for (int kk = 0; kk < 32; kk += 4) {
      const int kp = (kk >> 1) + kpHalf;               // kk+khalf is even
      v2f a[2], b[4];
      #pragma unroll
      for (int mt = 0; mt < 2; ++mt)
        a[mt] = *(const v2f*)&AsF[kp * KP_STRIDE + 2 * (mA0 + mt * 16)];
      #pragma unroll
      for (int nt = 0; nt < 4; ++nt)
        b[nt] = *(const v2f*)&BsF[kp * KP_STRIDE + 2 * (cB0 + nt * 16)];
      #pragma unroll
      for (int mt = 0; mt < 2; ++mt)
        #pragma unroll
        for (int nt = 0; nt < 4; ++nt)
          acc[mt][nt] = wmma_f32_16x16x4(a[mt], b[nt], acc[mt][nt]);
    }
    __syncthreads();
  }

  // Epilogue: R = pi_i*P_ij - pi_i*pi_j ; accumulate sum & sumsq in double
  double s = 0.0, ss = 0.0;
  const int mhalf = (lane >> 4) * 8;   // C layout: lanes 16-31 hold M+8
  #pragma unroll
  for (int mt = 0; mt < 2; ++mt) {
    #pragma unroll
    for (int nt = 0; nt < 4; ++nt) {
      int gj = bn * 128 + wn * 64 + nt * 16 + (lane & 15);
      float pj = pi[gj];
      #pragma unroll
      for (int v = 0; v < 8; ++v) {
        int gi = bm * 128 + wm * 32 + mt * 16 + mhalf + v;
        float pii = pi[gi];
        float p = acc[mt][nt][v];
        float R = pii * p - pii * pj;
        s  += (double)R;
        ss += (double)R * (double)R;
      }
    }
  }
  #pragma unroll
  for (int off = 16; off > 0; off >>= 1) {
    s  += __shfl_xor(s,  off, 32);
    ss += __shfl_xor(ss, off, 32);
  }
  if (lane == 0) { wred[wid] = s; wred[8 + wid] = ss; }
  __syncthreads();
  if (threadIdx.x == 0) {
    double ts = 0.0, tss = 0.0;
    for (int w = 0; w < 8; ++w) { ts += wred[w]; tss += wred[8 + w]; }
    int bid = blockIdx.y * gridDim.x + blockIdx.x;
    partials[2 * bid + 0] = ts;
    partials[2 * bid + 1] = tss;
  }
}

// ---------------------------------------------------------------------------
// K8: reduce partials -> stats[0]=mean, stats[1]=1/std (ddof=1), fixed order
// ---------------------------------------------------------------------------
__global__ __launch_bounds__(256) void finalize_kernel(
    const double* __restrict__ partials, int nb,
    float* __restrict__ stats, double NN)
{
  __shared__ double rs[256], rss[256];
  double s = 0.0, ss = 0.0;
  for (int i = threadIdx.x; i < nb; i += 256) {
    s  += partials[2 * i + 0];
    ss += partials[2 * i + 1];
  }
  rs[threadIdx.x] = s; rss[threadIdx.x] = ss;
  __syncthreads();
  for (int o = 128; o > 0; o >>= 1) {
    if (threadIdx.x < o) { rs[threadIdx.x] += rs[threadIdx.x + o];
                           rss[threadIdx.x] += rss[threadIdx.x + o]; }
    __syncthreads();
  }
  if (threadIdx.x == 0) {
    double S = rs[0], SS = rss[0];
    double mean = S / NN;
    double var  = (SS - S * S / NN) / (NN - 1.0);
    stats[0] = (float)mean;
    stats[1] = (float)(1.0 / sqrt(var));
  }
}

// ---------------------------------------------------------------------------
// K9: out[q] = (pi_i*P_ij - pi_i*pi_j - mean) * invstd, P_ij = M[i,:]·M[:,j]
// One wave per query; M (64 MB) is L2-resident so the strided column reads hit.
// ---------------------------------------------------------------------------
__global__ __launch_bounds__(256) void out_kernel(
    const float* __restrict__ M, const float* __restrict__ pi,
    const long long* __restrict__ edges, const float* __restrict__ stats,
    float* __restrict__ out, int n, int Q)
{
  const int wid = threadIdx.x >> 5, lane = threadIdx.x & 31;
  int q = blockIdx.x * 8 + wid;
  if (q >= Q) return;
  long long i = edges[2 * (size_t)q + 0];
  long long j = edges[2 * (size_t)q + 1];
  const float* mi = M + (size_t)i * n;
  float s = 0.f;
  for (int k = lane; k < n; k += 32) s += mi[k] * M[(size_t)k * n + j];
  #pragma unroll
  for (int off = 16; off > 0; off >>= 1) s += __shfl_xor(s, off, 32);
  if (lane == 0) {
    float pii = pi[i], pj = pi[j];
    float R = pii * s - pii * pj;
    out[q] = (R - stats[0]) * stats[1];
  }
}

// ---------------------------------------------------------------------------
extern "C" void kernel_launch(void* const* d_in, const int* in_sizes, int n_in,
                              void* d_out, int out_size, void* d_ws, size_t ws_size,
                              hipStream_t stream)
{
  const float*     X         = (const float*)d_in[0];
  const float*     A         = (const float*)d_in[1];
  const float*     S         = (const float*)d_in[2];
  const float*     mask      = (const float*)d_in[3];
  const long long* aug_edges = (const long long*)d_in[4];
  const long long* edges     = (const long long*)d_in[5];
  const float*     W1        = (const float*)d_in[6];
  const float*     b1        = (const float*)d_in[7];
  const float*     W2        = (const float*)d_in[8];
  const float*     b2        = (const float*)d_in[9];
  const float*     W3        = (const float*)d_in[10];
  const float*     b3        = (const float*)d_in[11];

  const int n = in_sizes[0] / 128;        // N = 4096 (X is N x 128)
  const int E = in_sizes[4] / 2;          // # augmented edges
  const int Q = in_sizes[5] / 2;          // # query edges (== out_size)
  const size_t nn = (size_t)n * n;

  char* ws = (char*)d_ws;
  size_t off = 0;
  float* M = (float*)(ws + off);           off += nn * sizeof(float);
  float* dvec = (float*)(ws + off);        off += (size_t)n * sizeof(float);
  float* pivec = (float*)(ws + off);       off += (size_t)n * sizeof(float);
  off = (off + 63) & ~(size_t)63;
  const int gb = n / 128;                  // 32
  const int nblocks = gb * gb;             // 1024
  double* partials = (double*)(ws + off);  off += (size_t)nblocks * 2 * sizeof(double);
  off = (off + 63) & ~(size_t)63;
  float* stats = (float*)(ws + off);       off += 64;
  float* vals = (float*)(ws + off);        off += (size_t)E * sizeof(float);
  (void)ws_size; (void)n_in;

  mlp_kernel<<<(E + 7) / 8, 256, 0, stream>>>(X, aug_edges, E, W1, b1, W2, b2, W3, b3, vals);
  build_kernel<<<(unsigned)((nn + 255) / 256), 256, 0, stream>>>(A, S, mask, M, n);
  scatter_kernel<<<(E + 255) / 256, 256, 0, stream>>>(aug_edges, vals, M, E, n);
  rowsum_kernel<<<n, 256, 0, stream>>>(M, dvec, n);
  pi_kernel<<<1, 1024, 0, stream>>>(dvec, pivec, n);
  normalize_kernel<<<(unsigned)((nn + 255) / 256), 256, 0, stream>>>(M, dvec, n);
  dim3 gg(gb, gb);
  gemm_stats_kernel<<<gg, 256, 0, stream>>>(M, pivec, partials, n);
  finalize_kernel<<<1, 256, 0, stream>>>(partials, nblocks, stats, (double)nn);
  out_kernel<<<(Q + 7) / 8, 256, 0, stream>>>(M, pivec, edges, stats, (float*)d_out, n, Q);
}